// Caps_BN_51410758533766
// MI455X (gfx1250) — compile-verified
//
#include <hip/hip_runtime.h>

#define EPS     1e-5f
#define NUM_C   16
#define NUM_D   32
#define NBATCH  64
#define CD      (NUM_C * NUM_D)      // 512 channels
#define HW      1024                 // 32*32
#define NCOLS   (NBATCH * HW)        // 65536 columns per capsule

typedef float v2f __attribute__((ext_vector_type(2)));
typedef float v8f __attribute__((ext_vector_type(8)));

// ---------------------------------------------------------------------------
// Workspace layout (floats):
//   [0,    512)  scale[ch]  = rsqrt(var+eps)
//   [512, 1024)  shift[ch]  = -mean * rsqrt(var+eps)
//   [1024,17408) Wp[c][o][i] = W[c][o][i] * scale[c*32+i]
//   [17408,17920) b2[c][o]  = sum_i W[c][o][i]*shift[c*32+i] + bias[c*32+o]
// ---------------------------------------------------------------------------

// Pass 1: per-channel biased mean/var over (N,H,W); emit scale/shift.
__global__ void __launch_bounds__(256)
bn_stats_kernel(const float* __restrict__ x,
                float* __restrict__ scale, float* __restrict__ shift) {
    const int ch  = blockIdx.x;      // 0..511
    const int tid = threadIdx.x;     // 0..255
    float s = 0.0f, sq = 0.0f;
    for (int n = 0; n < NBATCH; ++n) {
        const float4 v =
            ((const float4*)(x + (size_t)n * CD * HW + (size_t)ch * HW))[tid];
        s  += v.x + v.y + v.z + v.w;
        sq += v.x * v.x + v.y * v.y + v.z * v.z + v.w * v.w;
    }
    __shared__ float rs[256];
    __shared__ float rq[256];
    rs[tid] = s; rq[tid] = sq;
    __syncthreads();
    for (int off = 128; off > 0; off >>= 1) {
        if (tid < off) { rs[tid] += rs[tid + off]; rq[tid] += rq[tid + off]; }
        __syncthreads();
    }
    if (tid == 0) {
        const float mean = rs[0] * (1.0f / NCOLS);
        const float var  = rq[0] * (1.0f / NCOLS) - mean * mean;
        const float inv  = rsqrtf(var + EPS);
        scale[ch] = inv;
        shift[ch] = -mean * inv;
    }
}

// Pass 2: fold BN scale into weights, BN shift (+bias) into per-output bias.
__global__ void __launch_bounds__(1024)
fold_kernel(const float* __restrict__ W, const float* __restrict__ bias,
            const float* __restrict__ scale, const float* __restrict__ shift,
            float* __restrict__ Wp, float* __restrict__ b2) {
    const int c = blockIdx.x;          // 0..15
    const int tid = threadIdx.x;       // 0..1023
    const int o = tid >> 5, i = tid & 31;
    const float w = W[((size_t)c * NUM_D + o) * NUM_D + i];
    Wp[((size_t)c * NUM_D + o) * NUM_D + i] = w * scale[c * NUM_D + i];
    __shared__ float ss[NUM_D][NUM_D];
    ss[o][i] = w * shift[c * NUM_D + i];
    __syncthreads();
    if (i == 0) {
        float acc = bias[c * NUM_D + o];
        #pragma unroll
        for (int k = 0; k < NUM_D; ++k) acc += ss[o][k];
        b2[c * NUM_D + o] = acc;
    }
}

// Pass 3: per-capsule GEMM  out[o, m] = Wp[o, i] * x[i, m] + b2[o]
// via V_WMMA_F32_16X16X4_F32. One wave computes a 32(o) x 64(cols) tile.
__global__ void __launch_bounds__(256)
caps_gemm_kernel(const float* __restrict__ x,
                 const float* __restrict__ Wp, const float* __restrict__ b2,
                 float* __restrict__ out) {
    const int tid    = threadIdx.x;
    const int wave   = tid >> 5;
    const int lane   = tid & 31;
    const int lane16 = lane & 15;
    const int half   = lane >> 4;           // 0: lanes 0-15, 1: lanes 16-31
    const int c      = blockIdx.y;
    const int m_base = (blockIdx.x * 8 + wave) * 64;   // global column base
    const int n      = m_base >> 10;        // batch index (tiles never cross)
    const int hw0    = m_base & 1023;       // spatial base within image

    // A fragments (folded weights), 16x4 tiles:
    //   lane<16: M=lane holds K=0(v0),K=1(v1); lane>=16: M=lane-16 holds K=2,3
    v2f a[2][8];
    #pragma unroll
    for (int mt = 0; mt < 2; ++mt) {
        const float* wrow = Wp + ((size_t)(c * 32 + mt * 16 + lane16)) * 32;
        #pragma unroll
        for (int ks = 0; ks < 8; ++ks)
            a[mt][ks] = *(const v2f*)(wrow + ks * 4 + half * 2);
    }

    // Accumulators seeded with folded bias: C VGPR r = b2[o = mt*16 + half*8 + r]
    v8f acc[2][4];
    #pragma unroll
    for (int mt = 0; mt < 2; ++mt) {
        const float* bb = b2 + c * 32 + mt * 16 + half * 8;
        v8f cinit;
        #pragma unroll
        for (int r = 0; r < 8; ++r) cinit[r] = bb[r];
        #pragma unroll
        for (int nt = 0; nt < 4; ++nt) acc[mt][nt] = cinit;
    }

    // K loop over i = 0..31 in steps of 4 (fully unrolled).
    const float* xcap = x + ((size_t)n * CD + c * 32) * HW;   // x[n, c*32, 0]
    #pragma unroll
    for (int ks = 0; ks < 8; ++ks) {
        const int i0 = ks * 4 + half * 2;   // B: lane<16 holds K=0,1; else K=2,3
        v2f b[4];
        #pragma unroll
        for (int nt = 0; nt < 4; ++nt) {
            const float* p = xcap + (size_t)i0 * HW + hw0 + nt * 16 + lane16;
            b[nt][0] = p[0];
            b[nt][1] = p[HW];
        }
        #pragma unroll
        for (int mt = 0; mt < 2; ++mt)
            #pragma unroll
            for (int nt = 0; nt < 4; ++nt)
                acc[mt][nt] = __builtin_amdgcn_wmma_f32_16x16x4_f32(
                    /*neg_a=*/false, a[mt][ks],
                    /*neg_b=*/false, b[nt],
                    /*c_mod=*/(short)0, acc[mt][nt],
                    /*reuse_a=*/false, /*reuse_b=*/false);
    }

    // Store: D VGPR r -> row o = mt*16 + half*8 + r, col = lane16 within tile.
    // Non-temporal: keep x resident in the 192MB L2 instead of the output.
    float* ocap = out + ((size_t)n * CD + c * 32) * HW;
    #pragma unroll
    for (int mt = 0; mt < 2; ++mt) {
        #pragma unroll
        for (int nt = 0; nt < 4; ++nt) {
            const int col = hw0 + nt * 16 + lane16;
            #pragma unroll
            for (int r = 0; r < 8; ++r) {
                const int o = mt * 16 + half * 8 + r;
                __builtin_nontemporal_store(acc[mt][nt][r],
                                            ocap + (size_t)o * HW + col);
            }
        }
    }
}

extern "C" void kernel_launch(void* const* d_in, const int* in_sizes, int n_in,
                              void* d_out, int out_size, void* d_ws, size_t ws_size,
                              hipStream_t stream) {
    const float* x    = (const float*)d_in[0];   // (64, 512, 32, 32)
    const float* W    = (const float*)d_in[1];   // (16, 32, 32)
    const float* bias = (const float*)d_in[2];   // (512,)
    float* out = (float*)d_out;

    float* ws    = (float*)d_ws;
    float* scale = ws;            // 512
    float* shift = ws + 512;      // 512
    float* Wp    = ws + 1024;     // 16384
    float* b2    = ws + 17408;    // 512

    bn_stats_kernel<<<dim3(CD), dim3(256), 0, stream>>>(x, scale, shift);
    fold_kernel<<<dim3(NUM_C), dim3(1024), 0, stream>>>(W, bias, scale, shift, Wp, b2);
    caps_gemm_kernel<<<dim3(NCOLS / (8 * 64), NUM_C), dim3(256), 0, stream>>>(x, Wp, b2, out);
}